// QuantumCircuitLayer_26697516712197
// MI455X (gfx1250) — compile-verified
//
#include <hip/hip_runtime.h>
#include <cstdint>

#define BATCH       524288
#define NQ          32
#define NLAYERS     8
#define GATE_FLOATS (NLAYERS * NQ * 8)                 // 2048 floats: 8 per (layer,qubit) gate
#define STR_FLOATS  (NLAYERS * NQ * 2)                 // 512 floats: (1-s, s) pairs
#define TBL_FLOATS  (GATE_FLOATS + STR_FLOATS)         // 2560 floats = 10240 B
#define TBL_CHUNKS  (TBL_FLOATS / 4)                   // 640 x 16B chunks
#define MAIN_BLOCK  256

typedef float v4f __attribute__((ext_vector_type(4)));  // native vectors -> b128 + NT builtins
typedef float v2f __attribute__((ext_vector_type(2)));  // (re,im) pair -> v_pk_*_f32

// ---------------------------------------------------------------------------
// Kernel 1: fold RX*RY*RZ into one 2x2 complex gate per (layer,qubit), plus
// (1-sigmoid, sigmoid) strength pairs. 256 threads total; runs once per call.
// ---------------------------------------------------------------------------
__global__ __launch_bounds__(256) void qc_prep(const float* __restrict__ rot,
                                               const float* __restrict__ ent,
                                               float* __restrict__ tbl) {
  const int t = threadIdx.x;            // 0..255
  const int l = t >> 5, q = t & 31;
  const float rx = rot[(l * NQ + q) * 3 + 0];
  const float ry = rot[(l * NQ + q) * 3 + 1];
  const float rz = rot[(l * NQ + q) * 3 + 2];
  float pi_, pr, sy, cy, sx, cx;
  __sincosf(0.5f * rz, &pi_, &pr);      // phase p = pr + i*pi_
  __sincosf(0.5f * ry, &sy, &cy);
  __sincosf(0.5f * rx, &sx, &cx);
  // U = RX*RY*RZ:
  //   u00 = p*(cx*cy - i sx*sy)      u01 = conj(p)*(-cx*sy - i sx*cy)
  //   u10 = p*(cx*sy - i sx*cy)      u11 = conj(p)*( cx*cy + i sx*sy)
  const float A = cx * cy, B = sx * sy, C = cx * sy, D = sx * cy;
  float* g = tbl + (l * NQ + q) * 8;
  g[0] =  pr * A + pi_ * B;   // u00r
  g[1] =  pi_ * A - pr * B;   // u00i
  g[2] = -pr * C - pi_ * D;   // u01r
  g[3] =  pi_ * C - pr * D;   // u01i
  g[4] =  pr * C + pi_ * D;   // u10r
  g[5] =  pi_ * C - pr * D;   // u10i
  g[6] =  pr * A + pi_ * B;   // u11r
  g[7] =  pr * B - pi_ * A;   // u11i
  if (q < NQ - 1) {
    const float e = ent[l * (NQ - 1) + q];
    const float s = 1.0f / (1.0f + __expf(-e));
    float* sp = tbl + GATE_FLOATS + (l * NQ + q) * 2;
    sp[0] = 1.0f - s;
    sp[1] = s;
  }
}

// ---------------------------------------------------------------------------
// Kernel 2: one thread = one batch row; full 32-qubit complex state in VGPRs
// as (re,im) pairs so the backend emits packed v_pk_fma_f32 (op_sel/neg_lo).
// Gate table staged into LDS via gfx1250 async global->LDS b128 loads, with
// the staging latency hidden behind the x loads + init sin/cos.
// ---------------------------------------------------------------------------
__global__ __launch_bounds__(MAIN_BLOCK) void qc_main(const float* __restrict__ x,
                                                      const float* __restrict__ tbl,
                                                      float* __restrict__ out) {
  __shared__ __attribute__((aligned(16))) float lds[TBL_FLOATS];
  const int tid = threadIdx.x;

  // ---- issue async stage of the 10240-byte table into LDS (ASYNCcnt path) ----
  {
    const uint32_t lbase = (uint32_t)(uintptr_t)(void*)&lds[0];  // LDS byte offset
    const uint64_t gbase = (uint64_t)(uintptr_t)tbl;
    const uint32_t off0  = (uint32_t)tid * 16u;
    uint32_t laddr = lbase + off0;
    asm volatile("global_load_async_to_lds_b128 %0, %1, %2"
                 :: "v"(laddr), "v"(off0), "s"(gbase) : "memory");
    const uint32_t off1 = off0 + 256u * 16u;
    laddr = lbase + off1;
    asm volatile("global_load_async_to_lds_b128 %0, %1, %2"
                 :: "v"(laddr), "v"(off1), "s"(gbase) : "memory");
    if (tid < TBL_CHUNKS - 512) {                     // tail: 128 chunks
      const uint32_t off2 = off0 + 512u * 16u;
      laddr = lbase + off2;
      asm volatile("global_load_async_to_lds_b128 %0, %1, %2"
                   :: "v"(laddr), "v"(off2), "s"(gbase) : "memory");
    }
  }

  // ---- overlap: stream x (non-temporal) and run init sin/cos while DMA runs ----
  const long b = (long)blockIdx.x * MAIN_BLOCK + tid;
  v2f s0[NQ], s1[NQ];                     // complex amplitudes (re,im)
  const v4f* xv = (const v4f*)(x + b * NQ);
  float xq[NQ];
#pragma unroll
  for (int v = 0; v < NQ / 4; ++v) {
    const v4f t4 = __builtin_nontemporal_load(&xv[v]);
    xq[4 * v + 0] = t4.x; xq[4 * v + 1] = t4.y;
    xq[4 * v + 2] = t4.z; xq[4 * v + 3] = t4.w;
  }
#pragma unroll
  for (int q = 0; q < NQ; ++q) {
    float s_, c_;
    __sincosf(xq[q] * 1.5707963267948966f, &s_, &c_);
    s0[q] = (v2f){c_, 0.0f};
    s1[q] = (v2f){s_, 0.0f};
  }

  // ---- now wait for the table DMA and make it visible block-wide ----
  asm volatile("s_wait_asynccnt 0" ::: "memory");
  __syncthreads();

  for (int l = 0; l < NLAYERS; ++l) {
    // ---- single-qubit gates: 2x2 complex matvec per qubit (packed f32) ----
    const float* gl = lds + l * (NQ * 8);
#pragma unroll
    for (int q = 0; q < NQ; ++q) {
      const float4 ga = *(const float4*)(gl + q * 8);      // u00r,u00i,u01r,u01i
      const float4 gb = *(const float4*)(gl + q * 8 + 4);  // u10r,u10i,u11r,u11i
      const v2f a0 = s0[q], a1 = s1[q];
      const v2f a0s = a0.yx, a1s = a1.yx;                  // swapped (im,re)
      const v2f n00 = (v2f){-ga.y, ga.y};                  // +-Im(u00) -> neg_lo
      const v2f n01 = (v2f){-ga.w, ga.w};
      const v2f n10 = (v2f){-gb.y, gb.y};
      const v2f n11 = (v2f){-gb.w, gb.w};
      // n0 = u00*a0 + u01*a1 ; n1 = u10*a0 + u11*a1   (complex)
      s0[q] = ga.x * a0 + n00 * a0s + ga.z * a1 + n01 * a1s;
      s1[q] = gb.x * a0 + n10 * a0s + gb.z * a1 + n11 * a1s;
    }
    // ---- entanglement scan (sequential across qubits, per-row independent) ----
    const float* sl = lds + GATE_FLOATS + l * (NQ * 2);
    v2f c1 = s1[0];
    float control = __builtin_amdgcn_sqrtf(c1.x * c1.x + c1.y * c1.y);
#pragma unroll
    for (int q = 1; q < NQ; ++q) {
      const float2 sp = *(const float2*)(sl + (q - 1) * 2); // (1-s, s)
      const float om  = sp.x;
      const float tt  = sp.y * control;
      const v2f st0 = s0[q], st1 = s1[q];
      const v2f n0 = om * st0 + tt * st1;
      const v2f n1 = om * st1 + tt * st0;
      s0[q] = n0; s1[q] = n1;
      control = __builtin_amdgcn_sqrtf(n1.x * n1.x + n1.y * n1.y);
    }
  }

  // ---- output |a1|^2, 128-bit non-temporal stores ----
  v4f* ov = (v4f*)(out + b * NQ);
#pragma unroll
  for (int v = 0; v < NQ / 4; ++v) {
    v4f r;
    r.x = s1[4 * v + 0].x * s1[4 * v + 0].x + s1[4 * v + 0].y * s1[4 * v + 0].y;
    r.y = s1[4 * v + 1].x * s1[4 * v + 1].x + s1[4 * v + 1].y * s1[4 * v + 1].y;
    r.z = s1[4 * v + 2].x * s1[4 * v + 2].x + s1[4 * v + 2].y * s1[4 * v + 2].y;
    r.w = s1[4 * v + 3].x * s1[4 * v + 3].x + s1[4 * v + 3].y * s1[4 * v + 3].y;
    __builtin_nontemporal_store(r, &ov[v]);
  }
}

extern "C" void kernel_launch(void* const* d_in, const int* in_sizes, int n_in,
                              void* d_out, int out_size, void* d_ws, size_t ws_size,
                              hipStream_t stream) {
  const float* x   = (const float*)d_in[0];  // (BATCH, 32) f32
  const float* rot = (const float*)d_in[1];  // (8, 32, 3)  f32
  const float* ent = (const float*)d_in[2];  // (8, 31)     f32
  float* out = (float*)d_out;                // (BATCH, 32) f32
  float* tbl = (float*)d_ws;                 // 10240 B gate/strength table

  qc_prep<<<1, 256, 0, stream>>>(rot, ent, tbl);
  qc_main<<<BATCH / MAIN_BLOCK, MAIN_BLOCK, 0, stream>>>(x, tbl, out);
}